// SequenceRegressionDecoder_55336358643501
// MI455X (gfx1250) — compile-verified
//
#include <hip/hip_runtime.h>
#include <hip/hip_bf16.h>

// Problem dims (compile-time constants from the reference)
#define BATCH 64
#define SEQL  1024
#define DIM   1280
#define DIM4  (DIM / 4)      // 320 float4 per row
#define LCH   4              // L-chunks for deterministic split reduction
#define LSEG  (SEQL / LCH)

typedef __attribute__((ext_vector_type(2))) float v2f;
typedef __attribute__((ext_vector_type(8))) float v8f;

// ---------------------------------------------------------------------------
// Kernel 1: partial ragged sums, float4 (global_load_b128) path.
// grid = (B, LCH), block = 320 (one float4 column per thread -> one full
// 5120B row per block iteration, fully coalesced).
// Fixed accumulation order: (s0+s1)+(s2+s3) with row-phase accumulators.
// ---------------------------------------------------------------------------
__global__ void pool_partial_kernel(const float4* __restrict__ prev,   // [B, L, D/4]
                                    const int*    __restrict__ lengths,
                                    float4*       __restrict__ partial) { // [B, LCH, D/4]
    const int b  = blockIdx.x;
    const int lc = blockIdx.y;
    const int d4 = threadIdx.x;                 // 0..319

    const int plen = lengths[b] + 2;            // pool length, 2..1023
    const int l0 = lc * LSEG;
    int l1 = l0 + LSEG;
    if (l1 > plen) l1 = plen;

    const float4* base = prev + (size_t)b * SEQL * DIM4 + d4;

    float4 s0 = {0.f, 0.f, 0.f, 0.f};
    float4 s1 = {0.f, 0.f, 0.f, 0.f};
    float4 s2 = {0.f, 0.f, 0.f, 0.f};
    float4 s3 = {0.f, 0.f, 0.f, 0.f};

    int l = l0;
    for (; l + 4 <= l1; l += 4) {
        float4 r0 = base[(size_t)(l + 0) * DIM4];
        float4 r1 = base[(size_t)(l + 1) * DIM4];
        float4 r2 = base[(size_t)(l + 2) * DIM4];
        float4 r3 = base[(size_t)(l + 3) * DIM4];
        s0.x += r0.x; s0.y += r0.y; s0.z += r0.z; s0.w += r0.w;
        s1.x += r1.x; s1.y += r1.y; s1.z += r1.z; s1.w += r1.w;
        s2.x += r2.x; s2.y += r2.y; s2.z += r2.z; s2.w += r2.w;
        s3.x += r3.x; s3.y += r3.y; s3.z += r3.z; s3.w += r3.w;
    }
    for (; l < l1; ++l) {
        float4 r = base[(size_t)l * DIM4];
        s0.x += r.x; s0.y += r.y; s0.z += r.z; s0.w += r.w;
    }

    float4 out;
    out.x = (s0.x + s1.x) + (s2.x + s3.x);
    out.y = (s0.y + s1.y) + (s2.y + s3.y);
    out.z = (s0.z + s1.z) + (s2.z + s3.z);
    out.w = (s0.w + s1.w) + (s2.w + s3.w);
    partial[((size_t)(b * LCH + lc)) * DIM4 + d4] = out;
}

// ---------------------------------------------------------------------------
// Kernel 2: combine L-chunks, divide by pool_len -> embedding [B, D].
// grid = B, block = 320 (one float4 per thread).
// ---------------------------------------------------------------------------
__global__ void pool_reduce_kernel(const float4* __restrict__ partial,  // [B, LCH, D/4]
                                   const int*    __restrict__ lengths,
                                   float4*       __restrict__ emb) {    // [B, D/4]
    const int b  = blockIdx.x;
    const int d4 = threadIdx.x;
    const float inv = 1.0f / (float)(lengths[b] + 2);

    float4 s = {0.f, 0.f, 0.f, 0.f};
    #pragma unroll
    for (int lc = 0; lc < LCH; ++lc) {
        float4 p = partial[((size_t)(b * LCH + lc)) * DIM4 + d4];
        s.x += p.x; s.y += p.y; s.z += p.z; s.w += p.w;
    }
    float4 e = {s.x * inv, s.y * inv, s.z * inv, s.w * inv};
    emb[(size_t)b * DIM4 + d4] = e;
}

// ---------------------------------------------------------------------------
// Kernel 3: x = relu(emb @ W + bias) via V_WMMA_F32_16X16X4_F32.
// grid = (B/16, D/16), block = 32 (one wave per 16x16 output tile).
//
// A (16x4 f32) VGPR layout per ISA 7.12.2:
//   lanes 0-15  : M = lane,      a[0] = K=0, a[1] = K=1
//   lanes 16-31 : M = lane - 16, a[0] = K=2, a[1] = K=3
// B (4x16) / C,D (16x16): rows striped across lanes within a VGPR:
//   lanes 0-15  hold rows {v}      (B: K=v,   C: M=v)
//   lanes 16-31 hold rows {v+half} (B: K=v+2, C: M=v+8)
// ---------------------------------------------------------------------------
__global__ void dense_wmma_kernel(const float* __restrict__ emb,
                                  const float* __restrict__ W,      // [D, D] row-major (k, n)
                                  const float* __restrict__ bias,   // [D]
                                  float*       __restrict__ x) {    // [B, D]
    const int m0   = blockIdx.x * 16;
    const int n0   = blockIdx.y * 16;
    const int lane = threadIdx.x;        // 0..31, EXEC all ones
    const int half = lane >> 4;          // 0 or 1
    const int lid  = lane & 15;

    v8f c = {};

    const float* arow = emb + (size_t)(m0 + lid) * DIM;   // A row for this lane

    for (int k = 0; k < DIM; k += 4) {
        const int ka = k + half * 2;     // this lane's K pair: {ka, ka+1}

        v2f a;
        a.x = arow[ka + 0];
        a.y = arow[ka + 1];

        v2f bfrag;
        bfrag.x = W[(size_t)(ka + 0) * DIM + n0 + lid];
        bfrag.y = W[(size_t)(ka + 1) * DIM + n0 + lid];

        // 8 args: (neg_a, A, neg_b, B, c_mod, C, reuse_a, reuse_b)
        c = __builtin_amdgcn_wmma_f32_16x16x4_f32(
                /*neg_a=*/false, a, /*neg_b=*/false, bfrag,
                /*c_mod=*/(short)0, c, /*reuse_a=*/false, /*reuse_b=*/false);
    }

    // Write back with bias + ReLU.  c[v] = x[m0 + v + half*8][n0 + lid]
    #pragma unroll
    for (int v = 0; v < 8; ++v) {
        const int row = v + half * 8;
        const int col = lid;
        float val = c[v] + bias[n0 + col];
        val = fmaxf(val, 0.0f);
        x[(size_t)(m0 + row) * DIM + n0 + col] = val;
    }
}

// ---------------------------------------------------------------------------
// Kernel 4: out[b] = dot(x[b,:], cls_w) + cls_b.
// grid = B, block = 320: each thread owns exactly one float4 of the dot,
// then wave32 shuffle + 10-entry LDS reduction (fixed order -> deterministic).
// ---------------------------------------------------------------------------
__global__ void classifier_kernel(const float4* __restrict__ x,      // [B, D/4]
                                  const float4* __restrict__ cls_w,  // [D/4]
                                  const float*  __restrict__ cls_b,  // [1]
                                  float*        __restrict__ out) {  // [B]
    const int b = blockIdx.x;
    const int t = threadIdx.x;           // 0..319

    float4 xv = x[(size_t)b * DIM4 + t];
    float4 wv = cls_w[t];
    float acc = (xv.x * wv.x + xv.y * wv.y) + (xv.z * wv.z + xv.w * wv.w);

    // wave32 shuffle reduction
    #pragma unroll
    for (int off = 16; off > 0; off >>= 1)
        acc += __shfl_down(acc, off, 32);

    __shared__ float smem[10];
    if ((t & 31) == 0) smem[t >> 5] = acc;
    __syncthreads();

    if (t == 0) {
        float total = 0.f;
        #pragma unroll
        for (int w = 0; w < 10; ++w) total += smem[w];
        out[b] = total + cls_b[0];
    }
}

// ---------------------------------------------------------------------------
// Launcher
// ---------------------------------------------------------------------------
extern "C" void kernel_launch(void* const* d_in, const int* in_sizes, int n_in,
                              void* d_out, int out_size, void* d_ws, size_t ws_size,
                              hipStream_t stream) {
    const float* prev    = (const float*)d_in[0];  // [B, L, D]
    const int*   lengths = (const int*)  d_in[1];  // [B]
    const float* dense_w = (const float*)d_in[2];  // [D, D]
    const float* dense_b = (const float*)d_in[3];  // [D]
    const float* cls_w   = (const float*)d_in[4];  // [D, 1]
    const float* cls_b   = (const float*)d_in[5];  // [1]
    float*       out     = (float*)d_out;          // [B, 1] -> B floats

    // Workspace layout (floats):
    //   partial : B * LCH * D   = 327680
    //   emb     : B * D         =  81920
    //   x       : B * D         =  81920
    float* ws_partial = (float*)d_ws;
    float* ws_emb     = ws_partial + (size_t)BATCH * LCH * DIM;
    float* ws_x       = ws_emb     + (size_t)BATCH * DIM;

    // 1) partial ragged sums (b128 loads)
    dim3 g1(BATCH, LCH);
    pool_partial_kernel<<<g1, 320, 0, stream>>>(
        (const float4*)prev, lengths, (float4*)ws_partial);

    // 2) combine + mean
    pool_reduce_kernel<<<BATCH, 320, 0, stream>>>(
        (const float4*)ws_partial, lengths, (float4*)ws_emb);

    // 3) dense + relu via WMMA f32 16x16x4
    dim3 g3(BATCH / 16, DIM / 16);
    dense_wmma_kernel<<<g3, 32, 0, stream>>>(ws_emb, dense_w, dense_b, ws_x);

    // 4) classifier dot product
    classifier_kernel<<<BATCH, 320, 0, stream>>>(
        (const float4*)ws_x, (const float4*)cls_w, cls_b, out);
}